// C3_17892833755747
// MI455X (gfx1250) — compile-verified
//
#include <hip/hip_runtime.h>

// CDNA5 / gfx1250, wave32. Implicit-GEMM binarized 5x5 conv:
//   - TDM (tensor_load_to_lds) DMAs the padded input slab into LDS (TENSORcnt)
//   - V_WMMA_F32_16X16X4_F32, K-outer / 8 ow-tiles inner, 8 live accumulators:
//     A fragment + B offsets shared across tiles, per-tile shift folded into
//     the ds_load immediate offset, no back-to-back WMMA RAW hazards
//   - non-temporal stores keep the 192MB L2 for the reused activations

typedef float v2f  __attribute__((ext_vector_type(2)));
typedef float v8f  __attribute__((ext_vector_type(8)));
typedef unsigned int u32x4 __attribute__((ext_vector_type(4)));
typedef unsigned int u32x8 __attribute__((ext_vector_type(8)));

#define IC     6
#define OC     16
#define IH     128
#define IW     128
#define OH     124
#define OW     124
#define KTAPS  25          // 5x5
#define KDIM   150         // 6*25
#define KPAD   152         // pad to multiple of 4 (38 K-steps)
#define KSTEP  38
#define NTILE  8           // 8 ow-tiles of 16 columns
#define TH     8           // output rows per block
#define LROWS  (TH + 4)    // 12 resident input rows (always exactly 12)
#define LPITCH 132         // 128 cols + 4 pad DWORDs (TDM pad feature)

// Compile-time LDS offset for reduction index k = c*25 + kh*5 + kw.
__device__ constexpr int koff(int k) {
    return (k < KDIM)
        ? ((k / KTAPS) * LROWS + (k % KTAPS) / 5) * LPITCH + (k % KTAPS) % 5
        : 0;
}

__global__ __launch_bounds__(256)
void c3_bin_conv_wmma(const float* __restrict__ x,
                      const float* __restrict__ W,
                      float* __restrict__ out)
{
    __shared__ float xs[IC * LROWS * LPITCH];  // 9504 f32 = 38 KB staged slab
    __shared__ float ws[OC * KPAD];            // binarized weights [oc][k]

    const int tid = threadIdx.x;
    const int n   = blockIdx.y;
    int ohBase = blockIdx.x * TH;
    if (ohBase > IH - LROWS) ohBase = IH - LROWS;   // last slab -> 116 (overlap ok)

    // ---- TDM: one DMA of x[n][0..5][ohBase..ohBase+11][0..127] into LDS.
    // 3D tile: dim0 = 128 cols (f32), dim1 = 12 rows, dim2 = 6 channels.
    // Pad: +4 DWORDs after every 128 DWORDs -> row pitch 132 floats in LDS.
    if (tid == 0) {
        const unsigned lds_base = (unsigned)(size_t)(void*)xs;   // LDS byte offset
        const unsigned long long ga =
            (unsigned long long)(size_t)(x + ((size_t)n * IC * IH + ohBase) * IW);

        u32x4 g0;
        g0[0] = 1u;                                   // count=1, user descriptor
        g0[1] = lds_base;                             // lds_addr (bytes)
        g0[2] = (unsigned)ga;                         // global_addr[31:0]
        g0[3] = (unsigned)((ga >> 32) & 0x1FFFFFFull) // global_addr[56:32]
              | (2u << 30);                           // type = 2 ("image")

        u32x8 g1;
        g1[0] = (2u << 16)        // data_size = 4B
              | (1u << 20)        // pad_enable
              | (6u << 22)        // pad_interval: 128 DWORDs
              | (3u << 25);       // pad_amount: 4 DWORDs
        g1[1] = (unsigned)IW << 16;          // tensor_dim0[15:0] @ bits 63:48
        g1[2] = (unsigned)IH << 16;          // dim0 hi=0 | tensor_dim1[15:0]
        g1[3] = (unsigned)IW << 16;          // dim1 hi=0 | tile_dim0 = 128
        g1[4] = (unsigned)LROWS | (unsigned)IC << 16;  // tile_dim1=12 | tile_dim2=6
        g1[5] = (unsigned)IW;                // tensor_dim0_stride = 128 elems
        g1[6] = (unsigned)(IH * IW) << 16;   // stride0 hi=0 | tensor_dim1_stride lo
        g1[7] = 0u;                          // tensor_dim1_stride hi (16384>>16 = 0)

        u32x4 g2;                            // 3D: tensor_dim2 = 6, rest unused
        g2[0] = (unsigned)IC;
        g2[1] = 0u; g2[2] = 0u; g2[3] = 0u;
        u32x4 g3;
        g3[0] = 0u; g3[1] = 0u; g3[2] = 0u; g3[3] = 0u;

        asm volatile("tensor_load_to_lds %0, %1, %2, %3"
                     :: "s"(g0), "s"(g1), "s"(g2), "s"(g3)
                     : "memory");
    }

    // ---- overlap with the DMA: binarize weights into LDS
    // wb = sign(W); masked channels are already 0 -> sign(0)=0. k-pad -> 0.
    for (int i = tid; i < OC * KPAD; i += 256) {
        int oc = i / KPAD, k = i - oc * KPAD;
        float v = 0.0f;
        if (k < KDIM) {
            float w = W[oc * KDIM + k];
            v = (w > 0.0f) ? 1.0f : ((w < 0.0f) ? -1.0f : 0.0f);
        }
        ws[i] = v;
    }

    __builtin_amdgcn_s_wait_tensorcnt(0);   // TDM slab resident (no-op for waves 1..7)
    __syncthreads();

    // ---- compute: one output row per wave (8 waves); all 8 ow-tiles live at once
    const int wave = tid >> 5;
    const int lane = tid & 31;
    const int mj   = lane & 15;   // A row (M=oc) / B,D column (N)
    const int half = lane >> 4;   // K-half selector per 32-bit WMMA VGPR layout
    const int oh   = ohBase + wave;   // always < OH by construction (ohBase <= 116)

    // A: lane reads w[mj][4t + 2*half .. +1] once per K-step (ds_load_b64, imm offset)
    const float* wrowh = &ws[mj * KPAD + 2 * half];
    const float* bbase = &xs[wave * LPITCH + mj];

    v8f acc[NTILE];
    #pragma unroll
    for (int q = 0; q < NTILE; ++q)
        acc[q] = (v8f){0.f, 0.f, 0.f, 0.f, 0.f, 0.f, 0.f, 0.f};

    #pragma unroll
    for (int t = 0; t < KSTEP; ++t) {
        v2f a = *(const v2f*)(wrowh + 4 * t);
        // B offsets: compile-time pair selected once per step by lane half
        const int o0 = half ? koff(4 * t + 2) : koff(4 * t + 0);
        const int o1 = half ? koff(4 * t + 3) : koff(4 * t + 1);
        const float* p0 = bbase + o0;
        const float* p1 = bbase + o1;
        #pragma unroll
        for (int q = 0; q < NTILE; ++q) {
            v2f b;                       // per-tile +16 cols -> 64B imm ds offset
            b.x = p0[16 * q];
            b.y = p1[16 * q];
            acc[q] = __builtin_amdgcn_wmma_f32_16x16x4_f32(
                         false, a, false, b, (short)0, acc[q], false, false);
        }
    }

    // ---- store: D 16x16 f32 layout: VGPR v -> M = v + 8*half, N = lane&15
    #pragma unroll
    for (int q = 0; q < NTILE; ++q) {
        const int ow = 16 * q + mj;
        if (ow < OW) {                   // only masks lanes 12..15 of the last tile
            float* op = out + (((size_t)n * OC + 8 * half) * OH + oh) * OW + ow;
            #pragma unroll
            for (int v = 0; v < 8; ++v)
                __builtin_nontemporal_store(acc[q][v], op + (size_t)v * (OH * OW));
        }
    }
}

extern "C" void kernel_launch(void* const* d_in, const int* in_sizes, int n_in,
                              void* d_out, int out_size, void* d_ws, size_t ws_size,
                              hipStream_t stream) {
    (void)in_sizes; (void)n_in; (void)d_ws; (void)ws_size; (void)out_size;
    const float* x = (const float*)d_in[0];   // [256,6,128,128] f32
    const float* W = (const float*)d_in[1];   // [16,6,5,5] f32 (pre-masked)
    float* out = (float*)d_out;               // [256,16,124,124] f32

    dim3 grid((OH + TH - 1) / TH, 256);       // 16 oh-slabs x 256 batch
    c3_bin_conv_wmma<<<grid, 256, 0, stream>>>(x, W, out);
}